// Gridding_20486994002218
// MI455X (gfx1250) — compile-verified
//
#include <hip/hip_runtime.h>

#define GRID_D 128
#define GRID_V (GRID_D * GRID_D * GRID_D)

// No-return fp32 atomic add at device scope (L2 atomic units), with a
// compile-time instruction byte offset so the 8 corners of one point share a
// single 64-bit address VGPR pair and need no per-corner VALU address math.
#define GATOMIC_OFF(p, v, OFF)                                                  \
    asm volatile("global_atomic_add_f32 %0, %1, off offset:" #OFF               \
                 " scope:SCOPE_DEV"                                             \
                 :: "v"(p), "v"(v) : "memory")

__device__ __forceinline__ void gatomic_add_f32(float* p, float v) {
    asm volatile("global_atomic_add_f32 %0, %1, off scope:SCOPE_DEV"
                 :: "v"(p), "v"(v) : "memory");
}

// Zero the output grid with wide (B128) stores; lines land in L2 so the
// subsequent atomic pass resolves entirely inside the 192MB L2.
__global__ void zero_f32(float* __restrict__ out, int n) {
    int i = blockIdx.x * blockDim.x + threadIdx.x;
    int n4 = n >> 2;
    if (i < n4) {
        *reinterpret_cast<float4*>(out + ((size_t)i << 2)) =
            float4{0.f, 0.f, 0.f, 0.f};
    }
    if (i == 0) {                       // tail (n % 4 elements, 0 here)
        for (int k = n4 << 2; k < n; ++k) out[k] = 0.f;
    }
}

// One thread per point. blockIdx.y = batch (no per-thread idiv).
__global__ void gridding_scatter(const float* __restrict__ pts,
                                 float* __restrict__ out,
                                 int n_per_batch) {
    int t = blockIdx.x * blockDim.x + threadIdx.x;
    if (t >= n_per_batch) return;
    size_t pt = (size_t)blockIdx.y * n_per_batch + t;

    // Stream ahead: speculative global_prefetch_b8 (OOB silently dropped).
    __builtin_prefetch(pts + 3 * pt + 4096, 0, 3);

    const float sf = 64.0f;                 // 128 / 2
    float px = pts[3 * pt + 0] * sf;
    float py = pts[3 * pt + 1] * sf;
    float pz = pts[3 * pt + 2] * sf;

    // padding filter: exactly-zero points carry zero weight -> skip entirely
    if (fabsf(px) + fabsf(py) + fabsf(pz) == 0.0f) return;

    const float lo = -64.0f;
    const float hi = 64.0f - 1e-5f;         // maxs + 1 - EPS, folded in f32
    px = fminf(fmaxf(px, lo), hi);
    py = fminf(fmaxf(py, lo), hi);
    pz = fminf(fmaxf(pz, lo), hi);

    float lx = floorf(px), ly = floorf(py), lz = floorf(pz);
    float fx = px - lx,    fy = py - ly,    fz = pz - lz;     // in [0,1)
    int   ix = (int)lx + 64, iy = (int)ly + 64, iz = (int)lz + 64;

    float w0x = 1.0f - fx, w1x = fx;
    float w0y = 1.0f - fy, w1y = fy;
    float w0z = 1.0f - fz, w1z = fz;

    float* base = out + (size_t)blockIdx.y * GRID_V
                      + ((ix * GRID_D + iy) * GRID_D + iz);

    if (ix < GRID_D - 1 && iy < GRID_D - 1 && iz < GRID_D - 1) {
        // Interior fast path (~97.7% of points): all 8 corners valid.
        // Weight op order matches the reference: (wx*wy)*wz.
        float wxy00 = w0x * w0y, wxy01 = w0x * w1y;
        float wxy10 = w1x * w0y, wxy11 = w1x * w1y;
        GATOMIC_OFF(base, wxy00 * w0z, 0);      // (0,0,0)
        GATOMIC_OFF(base, wxy00 * w1z, 4);      // (0,0,1)
        GATOMIC_OFF(base, wxy01 * w0z, 512);    // (0,1,0)
        GATOMIC_OFF(base, wxy01 * w1z, 516);    // (0,1,1)
        GATOMIC_OFF(base, wxy10 * w0z, 65536);  // (1,0,0)
        GATOMIC_OFF(base, wxy10 * w1z, 65540);  // (1,0,1)
        GATOMIC_OFF(base, wxy11 * w0z, 66048);  // (1,1,0)
        GATOMIC_OFF(base, wxy11 * w1z, 66052);  // (1,1,1)
    } else {
        // Boundary slow path: drop corners at index == GRID_D.
        #pragma unroll
        for (int i = 0; i < 2; ++i) {
            if (ix + i >= GRID_D) continue;     // lower bound always >= 0
            float wx = i ? w1x : w0x;
            #pragma unroll
            for (int j = 0; j < 2; ++j) {
                if (iy + j >= GRID_D) continue;
                float wxy = wx * (j ? w1y : w0y);
                #pragma unroll
                for (int k = 0; k < 2; ++k) {
                    if (iz + k >= GRID_D) continue;
                    gatomic_add_f32(
                        base + ((i * GRID_D + j) * GRID_D + k),
                        wxy * (k ? w1z : w0z));
                }
            }
        }
    }
}

extern "C" void kernel_launch(void* const* d_in, const int* in_sizes, int n_in,
                              void* d_out, int out_size, void* d_ws, size_t ws_size,
                              hipStream_t stream) {
    const float* pts = (const float*)d_in[0];
    float*       out = (float*)d_out;

    int total_pts   = in_sizes[0] / 3;           // B * N
    int B           = out_size / GRID_V;         // 4
    if (B < 1) B = 1;
    int n_per_batch = total_pts / B;

    // Pass 1: zero the grid (d_out is poisoned by the harness).
    int zthreads = (out_size + 3) / 4;
    zero_f32<<<(zthreads + 255) / 256, 256, 0, stream>>>(out, out_size);

    // Pass 2: trilinear scatter with device-scope fp32 L2 atomics.
    dim3 grid((n_per_batch + 255) / 256, B);
    gridding_scatter<<<grid, 256, 0, stream>>>(pts, out, n_per_batch);
}